// GCNConv_31404800868549
// MI455X (gfx1250) — compile-verified
//
#include <hip/hip_runtime.h>
#include <hip/hip_bf16.h>

typedef __attribute__((ext_vector_type(2))) float v2f;
typedef __attribute__((ext_vector_type(8))) float v8f;

#define IN_F  128
#define OUT_F 32

// -------------------------------------------------------------------------
// Kernel 1: h = x @ W  via V_WMMA_F32_16X16X4_F32 (exact fp32 matrix path)
// One wave -> one 16-row tile of h, full N=32 (two 16x16 accumulators).
// -------------------------------------------------------------------------
__global__ void gcn_gemm_wmma(const float* __restrict__ x,
                              const float* __restrict__ W,
                              float* __restrict__ h, int M)
{
    const int lane  = threadIdx.x & 31;
    const int wib   = threadIdx.x >> 5;
    const int tile  = blockIdx.x * (blockDim.x >> 5) + wib;
    const int row0  = tile * 16;
    if (row0 >= M) return;

    const int m     = lane & 15;          // A: row in tile; B/C/D: column
    const int khalf = (lane >> 4) << 1;   // 0 for lanes 0-15, 2 for lanes 16-31

    v8f acc0 = {};                        // columns 0..15
    v8f acc1 = {};                        // columns 16..31

    const float* xrow = x + (size_t)(row0 + m) * IN_F;

    #pragma unroll 4
    for (int k0 = 0; k0 < IN_F; k0 += 4) {
        // A fragment (16x4): this lane's row, K = k0+khalf, k0+khalf+1
        v2f a;
        a.x = xrow[k0 + khalf];
        a.y = xrow[k0 + khalf + 1];

        // B fragments (4x16): lane = column, K split across half-waves
        const float* wk = W + (size_t)(k0 + khalf) * OUT_F;
        v2f b0, b1;
        b0.x = wk[m];              b0.y = wk[OUT_F + m];
        b1.x = wk[16 + m];         b1.y = wk[OUT_F + 16 + m];

        acc0 = __builtin_amdgcn_wmma_f32_16x16x4_f32(
                   false, a, false, b0, (short)0, acc0, false, false);
        acc1 = __builtin_amdgcn_wmma_f32_16x16x4_f32(
                   false, a, false, b1, (short)0, acc1, false, false);
    }

    // D layout: VGPR r -> M = r (lanes 0-15) / r+8 (lanes 16-31), N = lane&15
    const int mbase = (lane >> 4) << 3;
    #pragma unroll
    for (int r = 0; r < 8; ++r) {
        float* hr = h + (size_t)(row0 + mbase + r) * OUT_F;
        hr[m]      = acc0[r];
        hr[16 + m] = acc1[r];
    }
}

// -------------------------------------------------------------------------
// Kernel 2: zero the output (harness poisons d_out; segment-sum needs zeros)
// -------------------------------------------------------------------------
__global__ void gcn_zero(float4* __restrict__ out, int n4)
{
    int i = blockIdx.x * blockDim.x + threadIdx.x;
    if (i < n4) out[i] = make_float4(0.f, 0.f, 0.f, 0.f);
}

// -------------------------------------------------------------------------
// Kernel 3: edge scatter. lane == feature (OUT_F == warpSize == 32).
// Each wave loads 32 edges coalesced, then per edge does one coalesced
// 128B gather of h[src] and one coalesced global_atomic_add_f32 to out[dst].
// -------------------------------------------------------------------------
__global__ void gcn_scatter(const int*   __restrict__ esrc,
                            const int*   __restrict__ edst,
                            const float* __restrict__ ew,
                            const float* __restrict__ h,
                            float*       __restrict__ out, int E)
{
    const int lane   = threadIdx.x & 31;
    const int wid    = (blockIdx.x * blockDim.x + threadIdx.x) >> 5;
    const int nwaves = (gridDim.x * blockDim.x) >> 5;

    for (int e0 = wid * 32; e0 < E; e0 += nwaves * 32) {
        const int me   = e0 + lane;
        const bool ok  = (me < E);
        const int   s  = ok ? esrc[me] : 0;   // coalesced edge-list loads
        const int   d  = ok ? edst[me] : 0;
        const float w  = ok ? ew[me]   : 0.f;

        const int cnt = (E - e0 < 32) ? (E - e0) : 32;
        for (int j = 0; j < cnt; ++j) {
            const int   sj = __shfl(s, j, 32);
            const int   dj = __shfl(d, j, 32);
            const float wj = __shfl(w, j, 32);
            const float v  = wj * h[(size_t)sj * OUT_F + lane]; // 128B coalesced (L2 hit)
            atomicAdd(&out[(size_t)dj * OUT_F + lane], v);       // global_atomic_add_f32
        }
    }
}

// -------------------------------------------------------------------------
extern "C" void kernel_launch(void* const* d_in, const int* in_sizes, int n_in,
                              void* d_out, int out_size, void* d_ws, size_t ws_size,
                              hipStream_t stream)
{
    const float* x    = (const float*)d_in[0];
    const float* W    = (const float*)d_in[1];
    const int*   esrc = (const int*)  d_in[2];
    const int*   edst = (const int*)  d_in[3];
    const float* ew   = (const float*)d_in[4];
    float*       out  = (float*)d_out;

    const int M = in_sizes[0] / IN_F;   // 100000 nodes
    const int E = in_sizes[2];          // 3.2M edges

    float* h = (float*)d_ws;            // 100000*32*4 = 12.8 MB scratch

    // 1) GEMM: one wave per 16-row tile, 8 waves per 256-thread block
    {
        const int tiles  = (M + 15) / 16;
        const int blocks = (tiles + 7) / 8;
        gcn_gemm_wmma<<<blocks, 256, 0, stream>>>(x, W, h, M);
    }

    // 2) Zero output
    {
        const int n4     = out_size / 4;
        const int blocks = (n4 + 255) / 256;
        gcn_zero<<<blocks, 256, 0, stream>>>((float4*)out, n4);
    }

    // 3) Scatter-add over edges
    {
        gcn_scatter<<<1280, 256, 0, stream>>>(esrc, edst, ew, h, out, E);
    }
}